// GraphAttentionLayer_43568148251355
// MI455X (gfx1250) — compile-verified
//
#include <hip/hip_runtime.h>
#include <math.h>

#define IN_F 128
#define OUT_F 128
#define HEADS 8
#define HEAD_DIM 16
#define EDGE_F 64

typedef __attribute__((ext_vector_type(2))) float v2f;
typedef __attribute__((ext_vector_type(8))) float v8f;

// Monotone encoding of float into uint so integer atomicMax == float max.
__device__ __forceinline__ unsigned enc_f32(float f) {
    unsigned u = __float_as_uint(f);
    return (u & 0x80000000u) ? ~u : (u | 0x80000000u);
}
__device__ __forceinline__ float dec_f32(unsigned u) {
    return __uint_as_float((u & 0x80000000u) ? (u ^ 0x80000000u) : ~u);
}

// -------------------------------------------------------------------------
// Init: zero agg + per-(node,head) sums, set per-(node,head) max to enc(-inf)
// -------------------------------------------------------------------------
__global__ __launch_bounds__(256) void gat_init_kernel(
    float* __restrict__ agg, float* __restrict__ nodeSum,
    unsigned* __restrict__ nodeMax, size_t nAgg, size_t nNH)
{
    size_t i = (size_t)blockIdx.x * blockDim.x + threadIdx.x;
    if (i < nAgg) agg[i] = 0.0f;
    if (i < nNH) {
        nodeSum[i] = 0.0f;
        nodeMax[i] = 0x007FFFFFu; // enc(-inf)
    }
}

// -------------------------------------------------------------------------
// O[N,128] = X[N,128] @ W[128,128] + bias, via V_WMMA_F32_16X16X4_F32.
// Block = (32,8): 8 waves, wave y owns 16-col tile y; blockIdx.x = 16-row tile.
// A layout (16x4 f32): lane<16 -> M=lane, K={k0,k0+1}; lane>=16 -> K={k0+2,k0+3}.
// B layout (4x16 f32): v0 holds K rows {k0 | k0+2} striped over lanes (N=lane&15),
//                      v1 holds rows {k0+1 | k0+3}.
// C/D: lane<16 -> vgpr r = C[M=r][N=lane]; lane>=16 -> C[M=r+8][N=lane-16].
// -------------------------------------------------------------------------
__global__ __launch_bounds__(256) void gemm128_wmma_kernel(
    const float* __restrict__ X, const float* __restrict__ W,
    const float* __restrict__ bias, float* __restrict__ O, int N)
{
    const int rowTile = blockIdx.x;
    const int colTile = threadIdx.y;   // 0..7
    const int lane    = threadIdx.x;   // 0..31 (wave32)
    const int half    = lane >> 4;     // 0|1
    const int l16     = lane & 15;

    int row  = rowTile * 16 + l16;
    int rowc = row < N ? row : N - 1;            // clamp loads; EXEC stays all-1s
    const int col = colTile * 16 + l16;

    const float* __restrict__ arow = X + (size_t)rowc * IN_F;
    const float bb = bias[col];

    v8f acc = {};
    #pragma unroll 4
    for (int k0 = 0; k0 < IN_F; k0 += 4) {
        // A: two consecutive f32 per lane -> single global_load_b64
        v2f a = *(const v2f*)(arow + k0 + 2 * half);
        // B: two rows of W, 512B apart
        v2f b;
        b.x = W[(size_t)(k0 + 2 * half    ) * OUT_F + col];
        b.y = W[(size_t)(k0 + 2 * half + 1) * OUT_F + col];
        acc = __builtin_amdgcn_wmma_f32_16x16x4_f32(
            /*neg_a=*/false, a, /*neg_b=*/false, b,
            /*c_mod=*/(short)0, acc, /*reuse_a=*/false, /*reuse_b=*/false);
    }

    const int rbase = rowTile * 16 + half * 8;
    float* __restrict__ op = O + (size_t)rbase * OUT_F + col;
    if (rowTile * 16 + 16 <= N) {
        // Full tile (uniform branch): 8 plain stores with immediate row offsets.
        #pragma unroll
        for (int r = 0; r < 8; ++r) op[r * OUT_F] = acc[r] + bb;
    } else {
        // Tail tile: per-row guard.
        #pragma unroll
        for (int r = 0; r < 8; ++r)
            if (rbase + r < N) op[r * OUT_F] = acc[r] + bb;
    }
}

// -------------------------------------------------------------------------
// Pass 1: per (edge, head) score = (q[t,h]·k[s,h]) * 1/sqrt(16)
//         + edge_features[e] @ We[:,h] + be[h];  segment-max via atomicMax.
// -------------------------------------------------------------------------
__global__ __launch_bounds__(256) void edge_scores_kernel(
    const int* __restrict__ edge_index,      // [2,E]
    const float* __restrict__ edge_features, // [E,64]
    const float* __restrict__ We,            // [64,8]
    const float* __restrict__ be,            // [8]
    const float* __restrict__ q, const float* __restrict__ k,
    float* __restrict__ scores, unsigned* __restrict__ nodeMax, int E)
{
    int idx = blockIdx.x * blockDim.x + threadIdx.x;
    if (idx >= E * HEADS) return;
    const int e = idx >> 3;
    const int h = idx & 7;
    const int s = edge_index[e];
    const int t = edge_index[E + e];

    const float* __restrict__ qp = q + (size_t)t * OUT_F + h * HEAD_DIM;
    const float* __restrict__ kp = k + (size_t)s * OUT_F + h * HEAD_DIM;
    float dot = 0.0f;
    #pragma unroll
    for (int d = 0; d < HEAD_DIM; ++d) dot += qp[d] * kp[d];

    float b = be[h];
    const float* __restrict__ ef = edge_features + (size_t)e * EDGE_F;
    #pragma unroll 8
    for (int c = 0; c < EDGE_F; ++c) b += ef[c] * We[c * HEADS + h];

    const float sc = dot * 0.25f + b;   // HEAD_DIM^-0.5 = 0.25
    scores[idx] = sc;
    atomicMax(&nodeMax[(size_t)t * HEADS + h], enc_f32(sc));
}

// -------------------------------------------------------------------------
// Pass 2: e = exp(score - max[t,h]); overwrite scores with e; segment-sum.
// -------------------------------------------------------------------------
__global__ __launch_bounds__(256) void edge_exp_kernel(
    const int* __restrict__ edge_index,
    float* __restrict__ scores,
    const unsigned* __restrict__ nodeMax,
    float* __restrict__ nodeSum, int E)
{
    int idx = blockIdx.x * blockDim.x + threadIdx.x;
    if (idx >= E * HEADS) return;
    const int e = idx >> 3;
    const int h = idx & 7;
    const int t = edge_index[E + e];
    const float m  = dec_f32(nodeMax[(size_t)t * HEADS + h]);
    const float ev = expf(scores[idx] - m);
    scores[idx] = ev;
    atomicAdd(&nodeSum[(size_t)t * HEADS + h], ev);
}

// -------------------------------------------------------------------------
// Pass 3: prob = e / (sum + 1e-10); agg[t, h*16+d] += prob * v[s, h*16+d].
// -------------------------------------------------------------------------
__global__ __launch_bounds__(256) void aggregate_kernel(
    const int* __restrict__ edge_index,
    const float* __restrict__ scores,
    const float* __restrict__ nodeSum,
    const float* __restrict__ v,
    float* __restrict__ agg, int E)
{
    int idx = blockIdx.x * blockDim.x + threadIdx.x;
    if (idx >= E * HEADS) return;
    const int e = idx >> 3;
    const int h = idx & 7;
    const int s = edge_index[e];
    const int t = edge_index[E + e];
    const float prob = scores[idx] / (nodeSum[(size_t)t * HEADS + h] + 1e-10f);

    const float* __restrict__ vp = v + (size_t)s * OUT_F + h * HEAD_DIM;
    float* __restrict__ ap = agg + (size_t)t * OUT_F + h * HEAD_DIM;
    #pragma unroll
    for (int d = 0; d < HEAD_DIM; ++d) atomicAdd(&ap[d], prob * vp[d]);
}

// -------------------------------------------------------------------------
extern "C" void kernel_launch(void* const* d_in, const int* in_sizes, int n_in,
                              void* d_out, int out_size, void* d_ws, size_t ws_size,
                              hipStream_t stream)
{
    const float* X  = (const float*)d_in[0];
    const int*   EI = (const int*)  d_in[1];
    const float* EF = (const float*)d_in[2];
    const float* Wq = (const float*)d_in[3];
    const float* bq = (const float*)d_in[4];
    const float* Wk = (const float*)d_in[5];
    const float* bk = (const float*)d_in[6];
    const float* Wv = (const float*)d_in[7];
    const float* bv = (const float*)d_in[8];
    const float* We = (const float*)d_in[9];
    const float* be = (const float*)d_in[10];
    const float* Wo = (const float*)d_in[11];
    const float* bo = (const float*)d_in[12];
    float* out = (float*)d_out;

    const int N = in_sizes[0] / IN_F;
    const int E = in_sizes[1] / 2;

    // Workspace layout (floats): q | k | v | agg | scores | nodeSum | nodeMax
    float* ws = (float*)d_ws;
    const size_t NK = (size_t)N * OUT_F;
    float*    q       = ws;
    float*    k       = q + NK;
    float*    v       = k + NK;
    float*    agg     = v + NK;
    float*    scores  = agg + NK;
    float*    nodeSum = scores + (size_t)E * HEADS;
    unsigned* nodeMax = (unsigned*)(nodeSum + (size_t)N * HEADS);

    const size_t nNH = (size_t)N * HEADS;
    {
        size_t n = NK;                      // covers agg; nNH << NK
        int blocks = (int)((n + 255) / 256);
        gat_init_kernel<<<blocks, 256, 0, stream>>>(agg, nodeSum, nodeMax, NK, nNH);
    }

    const int rowTiles = (N + 15) / 16;
    const dim3 gblk(32, 8);                 // 8 waves: one per 16-col tile

    gemm128_wmma_kernel<<<rowTiles, gblk, 0, stream>>>(X, Wq, bq, q, N);
    gemm128_wmma_kernel<<<rowTiles, gblk, 0, stream>>>(X, Wk, bk, k, N);
    gemm128_wmma_kernel<<<rowTiles, gblk, 0, stream>>>(X, Wv, bv, v, N);

    const int eThreads = E * HEADS;
    const int eBlocks  = (eThreads + 255) / 256;
    edge_scores_kernel<<<eBlocks, 256, 0, stream>>>(EI, EF, We, be, q, k,
                                                    scores, nodeMax, E);
    edge_exp_kernel<<<eBlocks, 256, 0, stream>>>(EI, scores, nodeMax, nodeSum, E);
    aggregate_kernel<<<eBlocks, 256, 0, stream>>>(EI, scores, nodeSum, v, agg, E);

    gemm128_wmma_kernel<<<rowTiles, gblk, 0, stream>>>(agg, Wo, bo, out, N);
}